// Pairwise_Loss_23433341567081
// MI455X (gfx1250) — compile-verified
//
#include <hip/hip_runtime.h>

// Pairwise loss (flat-reshape "unfold" quirk preserved exactly).
// K=3, DIL=2, THR=0.9, N_CLASSES=4, H=W=1024, L=2^20.
//
// r[l, j] = patches[(9l+j) >> 20][(9l+j) & (L-1)]   (flat reshape, NOT transpose)
// node = col 5, edges = cols {0,1,2,3,5,6,7,8}
// flag = exp(-|n_i - e_i|/2) >= 0.9  <=>  |n_i - e_i| <= -2*ln(0.9)
// prob = e_m*n_m + (1-e_m)(1-n_m),  loss = -sum(flag*log(prob)) / (4L)
//
//   kernel 1: grid-stride over 4*2^20 pixels; gather 9 image taps; flags via
//             threshold (no exp); lazily load+sigmoid logits taps only where
//             flag is set; wave32 shuffle reduce + LDS -> d_ws[block].
//   kernel 2: one wave reduces 2048 block partials; exact f32 cross-lane sum
//             via V_WMMA_F32_16X16X4_F32 with B = ones.

#define L_SIZE   (1u << 20)
#define W_IMG    1024
#define NCLS     4
#define NBLOCKS  2048

typedef __attribute__((ext_vector_type(2))) float v2f;
typedef __attribute__((ext_vector_type(8))) float v8f;

// -2*ln(0.9), round-to-nearest f32
#define FLAG_THRESH 0.21072103f

__device__ __forceinline__ float sigmoidf_(float x) {
    return 1.0f / (1.0f + __expf(-x));
}

__global__ __launch_bounds__(256) void pairwise_loss_partial(
    const float* __restrict__ logits, const float* __restrict__ image,
    float* __restrict__ partials)
{
    const unsigned tid    = blockIdx.x * blockDim.x + threadIdx.x;
    const unsigned stride = gridDim.x * blockDim.x;
    const unsigned NP     = NCLS * L_SIZE;

    float acc = 0.0f;  // sum of flag * log(prob)  (log(prob) <= 0)

    for (unsigned g = tid; g < NP; g += stride) {
        const unsigned c = g >> 20;
        const unsigned l = g & (L_SIZE - 1u);
        const float* __restrict__ img = image  + (size_t)c * L_SIZE;
        const float* __restrict__ lg  = logits + (size_t)c * L_SIZE;

        const unsigned t0 = 9u * l;

        float    ei[9];
        int      idxs[9];
        unsigned okmask = 0u;
#pragma unroll
        for (int j = 0; j < 9; ++j) {
            const unsigned t = t0 + (unsigned)j;
            const unsigned p = t >> 20;            // which of the 9 shifted copies
            const unsigned m = t & (L_SIZE - 1u);  // flat position inside it
            const int row = (int)(m >> 10);
            const int col = (int)(m & (W_IMG - 1u));
            const int di  = (int)(p / 3u) * 2 - 2; // {-2,0,2}
            const int dj  = (int)(p % 3u) * 2 - 2;
            const int rr  = row + di;
            const int cc  = col + dj;
            const bool ok = ((unsigned)rr < 1024u) && ((unsigned)cc < 1024u);
            const int idx = (int)m + (di << 10) + dj;  // == rr*1024 + cc
            ei[j]   = ok ? img[idx] : 0.0f;            // zero-pad
            idxs[j] = idx;
            okmask |= (ok ? 1u : 0u) << j;
        }

        const float ni = ei[5];  // node column = s//2 + 1 = 5
        // node sigmoid always needed: the j==5 edge term has |d|==0 -> flag==1
        const float nm = ((okmask >> 5) & 1u) ? sigmoidf_(lg[idxs[5]]) : 0.0f;

#pragma unroll
        for (int j = 0; j < 9; ++j) {
            if (j == 4) continue;  // edge cols = {0,1,2,3,5,6,7,8}
            if (__builtin_fabsf(ni - ei[j]) <= FLAG_THRESH) {
                // sigmoid applied BEFORE zero-padding in the reference
                const float emj = ((okmask >> j) & 1u)
                                      ? sigmoidf_(lg[idxs[j]]) : 0.0f;
                const float prob = emj * nm + (1.0f - emj) * (1.0f - nm);
                acc += __logf(prob);   // prob > 0 always here
            }
        }
    }

    // wave32 reduction (ds_bpermute path)
#pragma unroll
    for (int off = 16; off > 0; off >>= 1)
        acc += __shfl_down(acc, off, 32);

    __shared__ float wsum[8];
    const int lane = threadIdx.x & 31;
    const int wid  = threadIdx.x >> 5;
    if (lane == 0) wsum[wid] = acc;
    __syncthreads();
    if (wid == 0) {
        float v = (lane < 8) ? wsum[lane] : 0.0f;
#pragma unroll
        for (int off = 4; off > 0; off >>= 1)
            v += __shfl_down(v, off, 32);
        if (lane == 0) partials[blockIdx.x] = v;
    }
}

// One wave (32 lanes, EXEC all ones) reduces the block partials.
// Exact f32 cross-lane sum via V_WMMA_F32_16X16X4_F32 with B = ones:
//   A (16x4 f32): lanes 0-15 hold K=0 (vgpr0), K=1 (vgpr1); lanes 16-31 K=2,3.
//   a = (lanesum, 0)  =>  D[m][n] = lanesum[m] + lanesum[m+16].
//   Each lane sums its 8 D vgprs; lanes 0-15 get sum(rows 0..7),
//   lanes 16-31 get sum(rows 8..15); total = s8 + shfl(s8, 16).
__global__ __launch_bounds__(32) void pairwise_loss_final(
    const float* __restrict__ partials, int nPartials, float* __restrict__ out)
{
    const int lane = threadIdx.x;
    float s = 0.0f;
    for (int i = lane; i < nPartials; i += 32)
        s += partials[i];

    v2f a; a.x = s;    a.y = 0.0f;
    v2f b; b.x = 1.0f; b.y = 1.0f;
    v8f cacc = {};
    v8f d = __builtin_amdgcn_wmma_f32_16x16x4_f32(
        /*neg_a=*/false, a, /*neg_b=*/false, b,
        /*c_mod=*/(short)0, cacc, /*reuse_a=*/false, /*reuse_b=*/false);

    float s8 = d[0] + d[1] + d[2] + d[3] + d[4] + d[5] + d[6] + d[7];
    const float total = s8 + __shfl(s8, 16, 32);

    if (lane == 0) {
        // per-class: -sum/L ; mean over 4 classes -> divide by 4L (exact pow2)
        out[0] = -total * (1.0f / (4.0f * 1048576.0f));
    }
}

extern "C" void kernel_launch(void* const* d_in, const int* in_sizes, int n_in,
                              void* d_out, int out_size, void* d_ws, size_t ws_size,
                              hipStream_t stream) {
    const float* logits = (const float*)d_in[0];  // inputs[:,0]  (mask logits)
    const float* image  = (const float*)d_in[1];  // image[:,0]
    float* out      = (float*)d_out;
    float* partials = (float*)d_ws;               // NBLOCKS floats (8 KB)

    pairwise_loss_partial<<<NBLOCKS, 256, 0, stream>>>(logits, image, partials);
    pairwise_loss_final<<<1, 32, 0, stream>>>(partials, NBLOCKS, out);
}